// MEFPN3D_73882027426149
// MI455X (gfx1250) — compile-verified
//
#include <hip/hip_runtime.h>

// ---------------- problem constants (from reference) ----------------
#define BATCHES 16
#define NPTS    100000
#define NBOX    64
#define KTOP    2048
#define TILES_PER_BATCH (NPTS / 16)          // 6250 exact
#define TPW     5                             // 16-point tiles per wave (6250 % 5 == 0)
#define WAVES_PER_BATCH (TILES_PER_BATCH/TPW) // 1250
#define R2      0.04f                         // (TENSOR_STRIDE*VOXEL_SIZE*GT_THRESHOLD)^2

typedef __attribute__((ext_vector_type(2))) float v2f;
typedef __attribute__((ext_vector_type(8))) float v8f;

// Monotonic float->uint key: larger float => larger key.
__device__ __forceinline__ unsigned fkey(float f) {
    unsigned u = __float_as_uint(f);
    return (u & 0x80000000u) ? ~u : (u | 0x80000000u);
}

// ================= Kernel 1: exact per-batch radix top-k threshold =================
// One block per batch. 4-pass byte radix select over monotonic keys.
__global__ void __launch_bounds__(1024) topk_thresh_kernel(
        const float* __restrict__ scores, unsigned* __restrict__ ws) {
    __shared__ unsigned hist[256];
    __shared__ unsigned sPrefix, sKrem;
    const int b = blockIdx.x;
    const float* s = scores + (size_t)b * NPTS;
    const int tid = threadIdx.x;

    if (tid == 0) { sPrefix = 0u; sKrem = KTOP; }
    __syncthreads();

    for (int pass = 0; pass < 4; ++pass) {
        const int shift = 24 - pass * 8;
        for (int i = tid; i < 256; i += blockDim.x) hist[i] = 0u;
        __syncthreads();
        const unsigned prefix = sPrefix;          // upper bytes fixed so far
        for (int i = tid; i < NPTS; i += blockDim.x) {
            unsigned u = fkey(s[i]);
            bool match = (pass == 0) ? true : ((u >> (shift + 8)) == prefix);
            if (match) atomicAdd(&hist[(u >> shift) & 0xFFu], 1u);
        }
        __syncthreads();
        if (tid == 0) {
            unsigned krem = sKrem, cum = 0u;
            int bin = 0;
            for (int i = 255; i >= 0; --i) {      // scan from largest byte down
                if (cum + hist[i] >= krem) { bin = i; break; }
                cum += hist[i];
            }
            sKrem = krem - cum;                   // count still needed inside chosen bin
            sPrefix = (prefix << 8) | (unsigned)bin;
        }
        __syncthreads();
    }
    if (tid == 0) {
        ws[b]      = sPrefix;   // threshold key T
        ws[16 + b] = sKrem;     // #equal-to-T elements admitted to top-K
        ws[32 + b] = 0u;        // tie counter for kernel 2
    }
}

// ================= Kernel 2: WMMA gt-mask + prune mask + BCE loss =================
// One wave32 per 5 consecutive 16-point tiles. For each tile:
//   D = A(16x4) x B(4x16) + C,  A=[px,py,pz,|p|^2], B=[-2cx,-2cy,-2cz,1],
//   C[m][n] = |c_n|^2 - R2   (threshold folded into the accumulator)
// => hit iff D < 0. B and C are built once per wave and reused across tiles.
__global__ void __launch_bounds__(256) prune_loss_kernel(
        const float* __restrict__ scores,
        const float* __restrict__ points,     // [B, NPTS, 3]
        const float* __restrict__ gtc,        // [B, NBOX, 3]
        float* __restrict__ outMask,          // [B, NPTS]  (1.0 / 0.0)
        float* __restrict__ outLoss,          // [B, NPTS]
        const unsigned* __restrict__ thr,     // [B]
        const unsigned* __restrict__ nEq,     // [B]
        unsigned* __restrict__ tieCnt) {      // [B]
    const int wave  = blockIdx.x * (blockDim.x >> 5) + (threadIdx.x >> 5);
    const int lane  = threadIdx.x & 31;
    const int batch = wave / WAVES_PER_BATCH;
    const int wIn   = wave - batch * WAVES_PER_BATCH;
    const int m     = lane & 15;
    const bool lo   = lane < 16;

    // ---- B tiles (lanes = N; VGPR0=K0|K2, VGPR1=K1|K3) and C accumulators
    // C/D layout: value depends only on N = lane%16 -> broadcast per-lane scalar
    // across all 8 C registers.
    v2f bt[4];
    v8f ct[4];
#pragma unroll
    for (int t = 0; t < 4; ++t) {
        const float* cc = gtc + ((size_t)batch * NBOX + t * 16 + m) * 3;
        const float cx = cc[0], cy = cc[1], cz = cc[2];
        const float c2 = fmaf(cx, cx, fmaf(cy, cy, cz * cz));
        v2f bv;
        bv.x = lo ? (-2.0f * cx) : (-2.0f * cz);  // K0 = -2cx | K2 = -2cz
        bv.y = lo ? (-2.0f * cy) : 1.0f;          // K1 = -2cy | K3 = 1
        bt[t] = bv;
        const float cm2 = c2 - R2;                // hit iff |p|^2-2p.c+|c|^2-R2 < 0
        v8f cv = {cm2, cm2, cm2, cm2, cm2, cm2, cm2, cm2};
        ct[t] = cv;
    }

    const unsigned T  = thr[batch];
    const unsigned NE = nEq[batch];

    int base = wIn * (16 * TPW);
#pragma unroll
    for (int tile = 0; tile < TPW; ++tile, base += 16) {
        // ---- A matrix (ISA 32-bit A 16x4: lanes 0-15 = M; VGPR0=K0|K2, VGPR1=K1|K3)
        const float* pp = points + ((size_t)batch * NPTS + base + m) * 3;
        const float x = pp[0], y = pp[1], z = pp[2];
        const float p2 = fmaf(x, x, fmaf(y, y, z * z));
        v2f a;
        a.x = lo ? x : z;     // K0 = px | K2 = pz
        a.y = lo ? y : p2;    // K1 = py | K3 = |p|^2

        // ---- 4x V_WMMA_F32_16X16X4_F32 with threshold pre-loaded in C
        v8f d0 = __builtin_amdgcn_wmma_f32_16x16x4_f32(false, a, false, bt[0],
                                                       (short)0, ct[0], false, false);
        v8f d1 = __builtin_amdgcn_wmma_f32_16x16x4_f32(false, a, false, bt[1],
                                                       (short)0, ct[1], false, false);
        v8f d2 = __builtin_amdgcn_wmma_f32_16x16x4_f32(false, a, false, bt[2],
                                                       (short)0, ct[2], false, false);
        v8f d3 = __builtin_amdgcn_wmma_f32_16x16x4_f32(false, a, false, bt[3],
                                                       (short)0, ct[3], false, false);

        // ---- min-reduce the 4 box tiles per register, then one ballot each.
        // C/D layout: VGPR r -> M=r (lanes 0-15), M=r+8 (lanes 16-31).
        unsigned rowhit = 0u;   // uniform; bit m = gt hit for point base+m
#pragma unroll
        for (int r = 0; r < 8; ++r) {
            const float e = fminf(fminf(d0[r], d1[r]), fminf(d2[r], d3[r]));
            const unsigned msk = __builtin_amdgcn_ballot_w32(e < 0.0f);
            if (msk & 0xFFFFu) rowhit |= (1u << r);
            if (msk >> 16)     rowhit |= (1u << (r + 8));
        }

        // ---- outputs: lanes 0-15 write the loss, lanes 16-31 write the prune mask
        const size_t gidx = (size_t)batch * NPTS + base + m;
        const float s = scores[gidx];
        const float tgt = ((rowhit >> m) & 1u) ? 1.0f : 0.0f;
        if (lo) {
            // BCE-with-logits (stable): max(s,0) - s*t + log1p(exp(-|s|))
            outLoss[gidx] = fmaxf(s, 0.0f) - s * tgt + log1pf(expf(-fabsf(s)));
        } else {
            const unsigned u = fkey(s);
            bool smask = (u > T);
            if (u == T) {
                unsigned old = atomicAdd(&tieCnt[batch], 1u);
                smask = (old < NE);
            }
            outMask[gidx] = (smask || (tgt > 0.5f)) ? 1.0f : 0.0f;
        }
    }
}

// ================================ launch ================================
extern "C" void kernel_launch(void* const* d_in, const int* in_sizes, int n_in,
                              void* d_out, int out_size, void* d_ws, size_t ws_size,
                              hipStream_t stream) {
    const float* scores = (const float*)d_in[0];   // [B, NPTS]
    const float* points = (const float*)d_in[1];   // [B, NPTS, 3]
    const float* gtc    = (const float*)d_in[2];   // [B, NBOX, 3]

    float* outMask = (float*)d_out;                        // first output (prune_mask)
    float* outLoss = outMask + (size_t)BATCHES * NPTS;     // second output (losses)

    unsigned* ws = (unsigned*)d_ws;   // [0..15]=T, [16..31]=nEq, [32..47]=tieCnt

    hipLaunchKernelGGL(topk_thresh_kernel, dim3(BATCHES), dim3(1024), 0, stream,
                       scores, ws);

    const int totalWaves = BATCHES * WAVES_PER_BATCH;   // 20000 waves, 5 tiles each
    const int blocks = totalWaves / 8;                  // 8 waves (256 thr) per block
    hipLaunchKernelGGL(prune_loss_kernel, dim3(blocks), dim3(256), 0, stream,
                       scores, points, gtc, outMask, outLoss,
                       ws, ws + 16, ws + 32);
}